// Sinkhorn_seq_29386166239349
// MI455X (gfx1250) — compile-verified
//
#include <hip/hip_runtime.h>

// Sinkhorn loss on MI455X (gfx1250, wave32).
// Key insight: reference returns only the LAST batch sample's loss, so we
// compute everything for b=7 only. fp32 throughout (problem is tiny and
// L2-resident; 23.3 TB/s HBM makes the 20-iteration normalization ~14us).
// Column sums are done with V_WMMA_F32_16X16X4_F32 (ones-matrix reduction).

#define NN 1024
#define CC 64
#define B7_OFF (7 * NN * CC)   // offset of batch sample 7 in [8,1024,64]

typedef __attribute__((ext_vector_type(2))) float v2f;
typedef __attribute__((ext_vector_type(8))) float v8f;

// ---------------------------------------------------------------------------
// softmax over the N axis (axis=1) for one column c of one tensor, b=7 only.
// grid.x = 128: blocks 0..63 -> y_s column c, blocks 64..127 -> y_t column c.
// ---------------------------------------------------------------------------
__global__ void softmax_col_kernel(const float* __restrict__ y_s,
                                   const float* __restrict__ y_t,
                                   float* __restrict__ p_s,
                                   float* __restrict__ p_t) {
  const int c   = blockIdx.x & 63;
  const int sel = blockIdx.x >> 6;
  const float* src = (sel ? y_t : y_s) + B7_OFF + c;
  float*       dst = (sel ? p_t : p_s) + c;
  const int t = threadIdx.x;                 // 0..255
  __shared__ float red[256];

  float v[4];
  float m = -3.402823466e38f;
#pragma unroll
  for (int k = 0; k < 4; ++k) {
    const int n = t + k * 256;
    v[k] = 0.5f * src[n * CC];               // y / T, T = 2.0
    m = fmaxf(m, v[k]);
  }
  red[t] = m;
  __syncthreads();
  for (int s = 128; s > 0; s >>= 1) {
    if (t < s) red[t] = fmaxf(red[t], red[t + s]);
    __syncthreads();
  }
  m = red[0];
  __syncthreads();

  float sum = 0.f;
#pragma unroll
  for (int k = 0; k < 4; ++k) {
    v[k] = expf(v[k] - m);
    sum += v[k];
  }
  red[t] = sum;
  __syncthreads();
  for (int s = 128; s > 0; s >>= 1) {
    if (t < s) red[t] += red[t + s];
    __syncthreads();
  }
  const float inv = 1.0f / red[0];
#pragma unroll
  for (int k = 0; k < 4; ++k) {
    const int n = t + k * 256;
    dst[n * CC] = v[k] * inv;
  }
}

// ---------------------------------------------------------------------------
// W[i,j] = sum_c |p_s[i,c] - p_t[j,c]|  (cdist p=1), P = exp(-W/0.1).
// LDS-tiled: each 256-thread block produces a 32x32 tile (4 outputs/thread).
// ---------------------------------------------------------------------------
__global__ void dist_kernel(const float* __restrict__ p_s,
                            const float* __restrict__ p_t,
                            float* __restrict__ W,
                            float* __restrict__ P) {
  __shared__ float xs[32 * 64];
  __shared__ float ys[32 * 64];
  const int i0 = blockIdx.y * 32;
  const int j0 = blockIdx.x * 32;
  const int t  = threadIdx.x;                // 0..255
  for (int e = t; e < 32 * 64; e += 256) {
    const int r = e >> 6, c = e & 63;
    xs[e] = p_s[(i0 + r) * CC + c];
    ys[e] = p_t[(j0 + r) * CC + c];
  }
  __syncthreads();
#pragma unroll
  for (int k = 0; k < 4; ++k) {
    const int idx = t + k * 256;
    const int i = idx >> 5, j = idx & 31;
    const float* xr = xs + i * 64;
    const float* yr = ys + j * 64;
    float acc = 0.f;
#pragma unroll
    for (int c = 0; c < 64; ++c) acc += fabsf(xr[c] - yr[c]);
    const size_t o = (size_t)(i0 + i) * NN + (j0 + j);
    W[o] = acc;
    P[o] = expf(-10.0f * acc);               // exp(-W / EPSILON), EPSILON=0.1
  }
}

// ---------------------------------------------------------------------------
// Row normalization: P[i,:] /= sum_j P[i,j].  One block per row, fused.
// ---------------------------------------------------------------------------
__global__ void rownorm_kernel(float* __restrict__ P) {
  float* row = P + (size_t)blockIdx.x * NN;
  const int t = threadIdx.x;
  __shared__ float red[256];
  float s = 0.f;
  for (int j = t; j < NN; j += 256) s += row[j];
  red[t] = s;
  __syncthreads();
  for (int st = 128; st > 0; st >>= 1) {
    if (t < st) red[t] += red[t + st];
    __syncthreads();
  }
  const float inv = 1.0f / red[0];
  for (int j = t; j < NN; j += 256) row[j] *= inv;
}

// ---------------------------------------------------------------------------
// Column sums via V_WMMA_F32_16X16X4_F32:  D = ones(16x4) * B(4x16) + C.
// One wave (blockDim=32) owns a 16-column strip; accumulates 256 WMMAs over
// the 1024 rows. B-layout (ISA 32-bit B 4x16): VGPR0 lanes0-15 = row k0,
// lanes16-31 = row k0+2; VGPR1 = rows k0+1 / k0+3; column = lane&15.
// D VGPR0 lanes 0-15 = (M=0, N=lane) = column sums.  EXEC all-1s: full wave,
// uniform control flow before the WMMA loop.
// ---------------------------------------------------------------------------
__global__ void colsum_wmma_kernel(const float* __restrict__ P,
                                   float* __restrict__ csum) {
  const int lane = threadIdx.x;              // 0..31
  const int l16  = lane & 15;
  const int hi   = lane >> 4;                // 0 or 1
  const int colbase = blockIdx.x * 16;
  const float* p0 = P + colbase + l16;

  v8f c = {};
  v2f a = {1.0f, 1.0f};                      // ones matrix (layout-invariant)
  for (int k0 = 0; k0 < NN; k0 += 4) {
    const int r = k0 + 2 * hi;
    v2f b;
    b.x = p0[(size_t)r * NN];
    b.y = p0[(size_t)(r + 1) * NN];
    c = __builtin_amdgcn_wmma_f32_16x16x4_f32(
        /*neg_a=*/false, a, /*neg_b=*/false, b,
        /*c_mod=*/(short)0, c, /*reuse_a=*/false, /*reuse_b=*/false);
  }
  if (lane < 16) csum[colbase + l16] = c[0];
}

// ---------------------------------------------------------------------------
// Column normalization divide: P[i,j] /= csum[j].  Fully coalesced.
// ---------------------------------------------------------------------------
__global__ void coldiv_kernel(float* __restrict__ P,
                              const float* __restrict__ csum) {
  const int idx = blockIdx.x * 256 + threadIdx.x;   // < 1024*1024
  P[idx] /= csum[idx & (NN - 1)];
}

// ---------------------------------------------------------------------------
// loss = sum(P * W): per-row partial (tree reduce, deterministic), then final.
// ---------------------------------------------------------------------------
__global__ void rowdot_kernel(const float* __restrict__ P,
                              const float* __restrict__ W,
                              float* __restrict__ part) {
  const size_t base = (size_t)blockIdx.x * NN;
  const int t = threadIdx.x;
  __shared__ float red[256];
  float s = 0.f;
  for (int j = t; j < NN; j += 256) s += P[base + j] * W[base + j];
  red[t] = s;
  __syncthreads();
  for (int st = 128; st > 0; st >>= 1) {
    if (t < st) red[t] += red[t + st];
    __syncthreads();
  }
  if (t == 0) part[blockIdx.x] = red[0];
}

__global__ void finalize_kernel(const float* __restrict__ part,
                                float* __restrict__ out) {
  const int t = threadIdx.x;
  __shared__ float red[256];
  float s = 0.f;
  for (int i = t; i < NN; i += 256) s += part[i];
  red[t] = s;
  __syncthreads();
  for (int st = 128; st > 0; st >>= 1) {
    if (t < st) red[t] += red[t + st];
    __syncthreads();
  }
  if (t == 0) out[0] = 1.0e-3f * red[0];     // SCALE = 0.001
}

// ---------------------------------------------------------------------------
extern "C" void kernel_launch(void* const* d_in, const int* in_sizes, int n_in,
                              void* d_out, int out_size, void* d_ws, size_t ws_size,
                              hipStream_t stream) {
  (void)in_sizes; (void)n_in; (void)out_size; (void)ws_size;
  const float* y_s = (const float*)d_in[0];
  const float* y_t = (const float*)d_in[1];

  // Workspace layout (floats): ~8.9 MB total.
  float* ws   = (float*)d_ws;
  float* p_s  = ws;                          // 1024*64
  float* p_t  = p_s + NN * CC;               // 1024*64
  float* W    = p_t + NN * CC;               // 1024*1024
  float* P    = W + (size_t)NN * NN;         // 1024*1024
  float* csum = P + (size_t)NN * NN;         // 1024
  float* part = csum + NN;                   // 1024

  softmax_col_kernel<<<128, 256, 0, stream>>>(y_s, y_t, p_s, p_t);
  dist_kernel<<<dim3(32, 32), 256, 0, stream>>>(p_s, p_t, W, P);

  for (int it = 0; it < 20; ++it) {
    rownorm_kernel<<<NN, 256, 0, stream>>>(P);
    colsum_wmma_kernel<<<NN / 16, 32, 0, stream>>>(P, csum);
    coldiv_kernel<<<(NN * NN) / 256, 256, 0, stream>>>(P, csum);
  }

  rowdot_kernel<<<NN, 256, 0, stream>>>(P, W, part);
  finalize_kernel<<<1, 256, 0, stream>>>(part, (float*)d_out);
}